// GraphIsomorphism_65197603553462
// MI455X (gfx1250) — compile-verified
//
#include <hip/hip_runtime.h>
#include <hip/hip_bf16.h>
#include <math.h>

// ---------------------------------------------------------------------------
// GIN layer on MI455X (gfx1250, wave32):
//   u = segment_sum(h[src], dst);  out = LN(relu(u@W1+b1)@W2+b2 + h)
//
// Phase 1: CSR bin-by-dst aggregation (replaces 205M f32 atomics with 3.2M
//          int atomics + conflict-free coalesced gathers).
// Phase 2: fused MLP + residual + LayerNorm, exact fp32 GEMMs via
//          V_WMMA_F32_16X16X4_F32 (codegen-confirmed last round).
// ---------------------------------------------------------------------------

#define N_NODES 100000
#define D_IN    128
#define D_H     512
#define N_EDGES 1600000
#define LN_EPS  1e-5f
#define NBLK    ((N_NODES + 255) / 256)   // 391 scan blocks

typedef __attribute__((ext_vector_type(2))) float v2f;
typedef __attribute__((ext_vector_type(8))) float v8f;

#define U_STRIDE 132   // 128 + 4 pad: A-frag LDS reads hit distinct banks
#define H_STRIDE 516   // 512 + 4 pad: same trick for the 16x512 hid tile

// ===================== Phase 1: CSR build + aggregate ======================

__global__ void gin_zero_counts(int* __restrict__ count, int n) {
    int i = blockIdx.x * blockDim.x + threadIdx.x;
    if (i < n) count[i] = 0;
}

__global__ void gin_hist(const int* __restrict__ dst, int* __restrict__ count,
                         int n_edges) {
    int i = blockIdx.x * blockDim.x + threadIdx.x;
    if (i < n_edges) atomicAdd(&count[dst[i]], 1);
}

// per-block sums of count[] -> partials[block]
__global__ void gin_block_sum(const int* __restrict__ count,
                              int* __restrict__ partials, int n) {
    __shared__ int sdata[256];
    int t = threadIdx.x;
    int i = blockIdx.x * 256 + t;
    sdata[t] = (i < n) ? count[i] : 0;
    __syncthreads();
#pragma unroll
    for (int o = 128; o > 0; o >>= 1) {
        if (t < o) sdata[t] += sdata[t + o];
        __syncthreads();
    }
    if (t == 0) partials[blockIdx.x] = sdata[0];
}

// serial exclusive scan of 391 partials (negligible work, 1 thread)
__global__ void gin_scan_partials(int* __restrict__ partials, int nblk) {
    if (blockIdx.x == 0 && threadIdx.x == 0) {
        int running = 0;
        for (int i = 0; i < nblk; ++i) {
            int v = partials[i];
            partials[i] = running;
            running += v;
        }
    }
}

// in-block exclusive scan + global offset -> offsets[], cursor[]
__global__ void gin_block_scan(const int* __restrict__ count,
                               const int* __restrict__ partials,
                               int* __restrict__ offsets,
                               int* __restrict__ cursor, int n) {
    __shared__ int sdata[256];
    int t = threadIdx.x;
    int i = blockIdx.x * 256 + t;
    int v = (i < n) ? count[i] : 0;
    sdata[t] = v;
    __syncthreads();
#pragma unroll
    for (int o = 1; o < 256; o <<= 1) {
        int x = (t >= o) ? sdata[t - o] : 0;
        __syncthreads();
        sdata[t] += x;
        __syncthreads();
    }
    if (i < n) {
        int excl = partials[blockIdx.x] + sdata[t] - v;
        offsets[i] = excl;
        cursor[i]  = excl;
    }
    if (i == 0) offsets[N_NODES] = N_EDGES;  // total is known exactly
}

__global__ void gin_scatter_idx(const int* __restrict__ src,
                                const int* __restrict__ dst,
                                int* __restrict__ cursor,
                                int* __restrict__ sorted_src, int n_edges) {
    int i = blockIdx.x * blockDim.x + threadIdx.x;
    if (i < n_edges) {
        int pos = atomicAdd(&cursor[dst[i]], 1);
        sorted_src[pos] = src[i];
    }
}

// One wave32 per node: lane l owns floats [4l,4l+4). Lanes cooperatively
// fetch up to 32 edge indices, broadcast each via shfl, and accumulate the
// 512B h-row as coalesced float4 gathers. No float atomics anywhere.
__global__ void gin_aggregate(const float* __restrict__ h,
                              const int* __restrict__ offsets,
                              const int* __restrict__ sorted_src,
                              float* __restrict__ u) {
    int node = (blockIdx.x * blockDim.x + threadIdx.x) >> 5;
    int lane = threadIdx.x & 31;
    if (node >= N_NODES) return;
    int beg = offsets[node];
    int end = offsets[node + 1];
    float4 acc = make_float4(0.f, 0.f, 0.f, 0.f);
    for (int base = beg; base < end; base += 32) {
        int idx = base + lane;
        int s = (idx < end) ? sorted_src[idx] : 0;
        int cnt = end - base;
        if (cnt > 32) cnt = 32;
        for (int j = 0; j < cnt; ++j) {
            int sj = __shfl(s, j, 32);
            const float4 hv =
                *(const float4*)(h + (size_t)sj * D_IN + lane * 4);
            acc.x += hv.x;
            acc.y += hv.y;
            acc.z += hv.z;
            acc.w += hv.w;
        }
    }
    *(float4*)(u + (size_t)node * D_IN + lane * 4) = acc;
}

// ============== Phase 2: fused MLP + residual + LayerNorm ==================
// One 16-row node tile per block; 256 threads = 8 waves.
//   stage 1: HID(16x512) = relu(U @ W1 + b1)   (wave w -> N cols [64w,64w+64))
//   stage 2: X(16x128)   = HID @ W2 + b2       (wave w -> N cols [16w,16w+16))
//   stage 3: out = LN(X + h) * gamma + beta    (wave w -> rows 2w, 2w+1)
__launch_bounds__(256, 2)
__global__ void gin_mlp_ln_kernel(const float* __restrict__ u,
                                  const float* __restrict__ W1,
                                  const float* __restrict__ b1,
                                  const float* __restrict__ W2,
                                  const float* __restrict__ b2,
                                  const float* __restrict__ gamma,
                                  const float* __restrict__ beta,
                                  const float* __restrict__ h,
                                  float* __restrict__ out) {
    __shared__ float sU[16 * U_STRIDE];   //  8448 B
    __shared__ float sH[16 * H_STRIDE];   // 33024 B
    __shared__ float sX[16 * U_STRIDE];   //  8448 B

    const int row0 = blockIdx.x * 16;
    const int t    = threadIdx.x;
    const int lane = t & 31;
    const int w    = t >> 5;

    // ---- load 16x128 U tile into LDS (512 float4 chunks, coalesced) ----
    for (int i = t; i < 512; i += 256) {
        int r  = i >> 5;
        int c  = (i & 31) * 4;
        float4 v = *(const float4*)(u + (size_t)(row0 + r) * D_IN + c);
        *(float4*)&sU[r * U_STRIDE + c] = v;
    }
    __syncthreads();

    // per-lane WMMA fragment coordinates (ISA 7.12.2):
    const int m   = lane & 15;          // A row
    const int nl  = lane & 15;          // B/D col within 16-wide tile
    const int off = (lane >> 4) * 2;    // k sub-offset {0,2}

    // ---- stage 1: U(16x128) @ W1(128x512), 4 output tiles per wave ----
    v8f acc[4] = {};
    for (int kb = 0; kb < D_IN; kb += 4) {
        v2f a = *(const v2f*)&sU[m * U_STRIDE + kb + off];
#pragma unroll
        for (int j = 0; j < 4; ++j) {
            int n = w * 64 + j * 16 + nl;
            v2f b;
            b.x = W1[(size_t)(kb + off)     * D_H + n];
            b.y = W1[(size_t)(kb + off + 1) * D_H + n];
            acc[j] = __builtin_amdgcn_wmma_f32_16x16x4_f32(
                false, a, false, b, (short)0, acc[j], false, false);
        }
    }
    // bias + ReLU -> LDS hid tile
#pragma unroll
    for (int j = 0; j < 4; ++j) {
        int n = w * 64 + j * 16 + nl;
        float bias = b1[n];
#pragma unroll
        for (int v = 0; v < 8; ++v) {
            int r = v + 8 * (lane >> 4);
            float val = acc[j][v] + bias;
            sH[r * H_STRIDE + n] = val > 0.f ? val : 0.f;
        }
    }
    __syncthreads();

    // ---- stage 2: HID(16x512) @ W2(512x128), one 16x16 tile per wave ----
    v8f acc2 = {};
    const int n2 = w * 16 + nl;
    for (int kb = 0; kb < D_H; kb += 4) {
        v2f a = *(const v2f*)&sH[m * H_STRIDE + kb + off];
        v2f b;
        b.x = W2[(size_t)(kb + off)     * D_IN + n2];
        b.y = W2[(size_t)(kb + off + 1) * D_IN + n2];
        acc2 = __builtin_amdgcn_wmma_f32_16x16x4_f32(
            false, a, false, b, (short)0, acc2, false, false);
    }
    {
        float bias = b2[n2];
#pragma unroll
        for (int v = 0; v < 8; ++v) {
            int r = v + 8 * (lane >> 4);
            sX[r * U_STRIDE + n2] = acc2[v] + bias;
        }
    }
    __syncthreads();

    // ---- stage 3: LayerNorm(X + h); wave w handles rows 2w and 2w+1 ----
#pragma unroll
    for (int rr = 0; rr < 2; ++rr) {
        int r = w * 2 + rr;
        int c = lane * 4;
        float4 hv = *(const float4*)(h + (size_t)(row0 + r) * D_IN + c);
        float4 xv = *(const float4*)&sX[r * U_STRIDE + c];
        float x0 = xv.x + hv.x;
        float x1 = xv.y + hv.y;
        float x2 = xv.z + hv.z;
        float x3 = xv.w + hv.w;
        float s  = x0 + x1 + x2 + x3;
        float ss = x0 * x0 + x1 * x1 + x2 * x2 + x3 * x3;
#pragma unroll
        for (int d = 16; d >= 1; d >>= 1) {
            s  += __shfl_xor(s,  d, 32);
            ss += __shfl_xor(ss, d, 32);
        }
        float mean = s * (1.f / 128.f);
        float var  = ss * (1.f / 128.f) - mean * mean;
        float rstd = rsqrtf(var + LN_EPS);
        float4 g  = *(const float4*)(gamma + c);
        float4 be = *(const float4*)(beta + c);
        float4 o;
        o.x = (x0 - mean) * rstd * g.x + be.x;
        o.y = (x1 - mean) * rstd * g.y + be.y;
        o.z = (x2 - mean) * rstd * g.z + be.z;
        o.w = (x3 - mean) * rstd * g.w + be.w;
        *(float4*)(out + (size_t)(row0 + r) * D_IN + c) = o;
    }
}

// ---------------------------------------------------------------------------
extern "C" void kernel_launch(void* const* d_in, const int* in_sizes, int n_in,
                              void* d_out, int out_size, void* d_ws, size_t ws_size,
                              hipStream_t stream) {
    const float* h     = (const float*)d_in[0];
    const float* W1    = (const float*)d_in[1];
    const float* b1    = (const float*)d_in[2];
    const float* W2    = (const float*)d_in[3];
    const float* b2    = (const float*)d_in[4];
    const float* gamma = (const float*)d_in[5];
    const float* beta  = (const float*)d_in[6];
    const int*   src   = (const int*)d_in[7];
    const int*   dst   = (const int*)d_in[8];
    float* out = (float*)d_out;

    // ---- workspace layout (~59 MB) ----
    float* u          = (float*)d_ws;                    // 12.8M floats
    int*   count      = (int*)(u + (size_t)N_NODES * D_IN);
    int*   offsets    = count + N_NODES;                 // N_NODES+1
    int*   cursor     = offsets + N_NODES + 1;
    int*   partials   = cursor + N_NODES;                // NBLK (<512)
    int*   sorted_src = partials + 512;                  // N_EDGES

    const int eblk = (N_EDGES + 255) / 256;   // 6250
    const int nblk = NBLK;                    // 391

    // ---- phase 1: CSR build ----
    hipLaunchKernelGGL(gin_zero_counts, dim3(nblk), dim3(256), 0, stream,
                       count, N_NODES);
    hipLaunchKernelGGL(gin_hist, dim3(eblk), dim3(256), 0, stream,
                       dst, count, N_EDGES);
    hipLaunchKernelGGL(gin_block_sum, dim3(nblk), dim3(256), 0, stream,
                       count, partials, N_NODES);
    hipLaunchKernelGGL(gin_scan_partials, dim3(1), dim3(1), 0, stream,
                       partials, nblk);
    hipLaunchKernelGGL(gin_block_scan, dim3(nblk), dim3(256), 0, stream,
                       count, partials, offsets, cursor, N_NODES);
    hipLaunchKernelGGL(gin_scatter_idx, dim3(eblk), dim3(256), 0, stream,
                       src, dst, cursor, sorted_src, N_EDGES);

    // ---- phase 1b: conflict-free aggregation, one wave per node ----
    {
        long long threads = (long long)N_NODES * 32;
        int blocks = (int)((threads + 255) / 256);       // 12500
        hipLaunchKernelGGL(gin_aggregate, dim3(blocks), dim3(256), 0, stream,
                           h, offsets, sorted_src, u);
    }

    // ---- phase 2: fused MLP + residual + LayerNorm (16-row tiles) ----
    hipLaunchKernelGGL(gin_mlp_ln_kernel, dim3(N_NODES / 16), dim3(256), 0,
                       stream, u, W1, b1, W2, b2, gamma, beta, h, out);
}